// CGCNN_55233279427066
// MI455X (gfx1250) — compile-verified
//
#include <hip/hip_runtime.h>

typedef float v2f __attribute__((ext_vector_type(2)));
typedef float v8f __attribute__((ext_vector_type(8)));

#define BN_EPS 1e-5f

__device__ __forceinline__ float sigmoidf_(float x) { return 1.f / (1.f + expf(-x)); }
__device__ __forceinline__ float softplusf_(float x) {
    if (x > 20.f) return x;
    return log1pf(expf(x));
}
__device__ __forceinline__ float siluf_(float x) { return x * sigmoidf_(x); }

// ---------------------------------------------------------------------------
// Generic zero
// ---------------------------------------------------------------------------
__global__ void zero_kernel(float* __restrict__ p, int n) {
    for (int i = blockIdx.x * blockDim.x + threadIdx.x; i < n; i += gridDim.x * blockDim.x)
        p[i] = 0.f;
}

// ---------------------------------------------------------------------------
// WMMA GEMM: rows x K  @  K x 64  (fp32, V_WMMA_F32_16X16X4_F32)
// CONV=1: A row = concat(v[src], v[dst], edge_feats) (K=169, padded to 172),
//         two weight matrices (Wm, Ws) -> ym, ys
// CONV=0: A row = node_feats row (K=92), one weight matrix -> ym
// Also accumulates per-output-feature sum / sum-of-squares into `stats`
// (layout: [sumM 64][sqM 64] and for CONV additionally [sumS 64][sqS 64]).
// rows must be a multiple of 16 (E=800000, N=50000 both are).
// ---------------------------------------------------------------------------
template <int CONV>
__global__ __launch_bounds__(256) void gemm_wmma_kernel(
    const float* __restrict__ vin,
    const float* __restrict__ efeat,
    const int* __restrict__ src,
    const int* __restrict__ dst,
    const float* __restrict__ Wm,   // [KREAL, 64] row-major
    const float* __restrict__ Ws,   // [KREAL, 64] (CONV only)
    float* __restrict__ ym,         // [rows, 64]
    float* __restrict__ ys,         // [rows, 64] (CONV only)
    float* __restrict__ stats,
    int rows)
{
    constexpr int KP    = CONV ? 172 : 92;
    constexpr int KREAL = CONV ? 169 : 92;
    constexpr int NW    = CONV ? 2 : 1;
    constexpr int NSTAT = NW * 128;

    __shared__ float lds_w[NW * 64 * KP];   // weights transposed: [path][n][k]
    __shared__ float lds_stats[NSTAT];

    const int tid = threadIdx.x;

    // Stage weights into LDS, transposed so a K-pair is a contiguous b64.
    for (int i = tid; i < NW * 64 * KP; i += 256) {
        int wsel = i / (64 * KP);
        int r    = i - wsel * 64 * KP;
        int n    = r / KP;
        int k    = r - n * KP;
        const float* W = (CONV && wsel) ? Ws : Wm;
        lds_w[i] = (k < KREAL) ? W[k * 64 + n] : 0.f;
    }
    for (int i = tid; i < NSTAT; i += 256) lds_stats[i] = 0.f;
    __syncthreads();

    const int lane = tid & 31;
    const int r16  = lane & 15;   // A row within tile / B column within ntile
    const int kp2  = lane >> 4;   // K half: lanes 0-15 hold K={0,1}, 16-31 K={2,3}
    const int gw     = blockIdx.x * 8 + (tid >> 5);
    const int nwaves = gridDim.x * 8;
    const int ntiles = rows >> 4;

    const v8f vzero = {0.f, 0.f, 0.f, 0.f, 0.f, 0.f, 0.f, 0.f};

    float smA[4] = {0, 0, 0, 0}, sqA[4] = {0, 0, 0, 0};
    float smB[4] = {0, 0, 0, 0}, sqB[4] = {0, 0, 0, 0};

    for (int tile = gw; tile < ntiles; tile += nwaves) {
        const int row = tile * 16 + r16;
        const float* pa0;
        const float* pa1 = nullptr;
        const float* pe  = nullptr;
        if (CONV) {
            pa0 = vin + (size_t)src[row] * 64;
            pa1 = vin + (size_t)dst[row] * 64;
            pe  = efeat + (size_t)row * 41;
        } else {
            pa0 = vin + (size_t)row * KREAL;
        }

        v8f cm[4], cs[4];
#pragma unroll
        for (int t = 0; t < 4; ++t) { cm[t] = vzero; cs[t] = vzero; }

#pragma unroll 1
        for (int ks = 0; ks < KP / 4; ++ks) {
            const int k0 = ks * 4 + kp2 * 2;
            v2f a;
            if (CONV) {
                // region branches never straddle a lane-half pair (boundaries 64/128/168 even)
                if (k0 < 64)        { a[0] = pa0[k0];       a[1] = pa0[k0 + 1]; }
                else if (k0 < 128)  { a[0] = pa1[k0 - 64];  a[1] = pa1[k0 - 63]; }
                else if (k0 < 168)  { a[0] = pe[k0 - 128];  a[1] = pe[k0 - 127]; }
                else                { a[0] = (k0 == 168) ? pe[40] : 0.f; a[1] = 0.f; }
            } else {
                a[0] = pa0[k0]; a[1] = pa0[k0 + 1];
            }
#pragma unroll
            for (int t = 0; t < 4; ++t) {
                const int nb = t * 16 + r16;
                v2f bm = *(const v2f*)&lds_w[nb * KP + k0];
                cm[t] = __builtin_amdgcn_wmma_f32_16x16x4_f32(
                            false, a, false, bm, (short)0, cm[t], false, false);
                if (CONV) {
                    v2f bs = *(const v2f*)&lds_w[64 * KP + nb * KP + k0];
                    cs[t] = __builtin_amdgcn_wmma_f32_16x16x4_f32(
                                false, a, false, bs, (short)0, cs[t], false, false);
                }
            }
        }

        // Store tile + accumulate per-column stats.
        // D layout: VGPR i, lanes 0-15 -> M=i, lanes 16-31 -> M=8+i; column N = lane%16.
#pragma unroll
        for (int t = 0; t < 4; ++t) {
            const int n = t * 16 + r16;
            float s = 0.f, q = 0.f;
#pragma unroll
            for (int i = 0; i < 8; ++i) {
                const int M  = i + kp2 * 8;
                const float val = cm[t][i];
                ym[(size_t)(tile * 16 + M) * 64 + n] = val;
                s += val; q += val * val;
            }
            smA[t] += s; sqA[t] += q;
            if (CONV) {
                s = 0.f; q = 0.f;
#pragma unroll
                for (int i = 0; i < 8; ++i) {
                    const int M  = i + kp2 * 8;
                    const float val = cs[t][i];
                    ys[(size_t)(tile * 16 + M) * 64 + n] = val;
                    s += val; q += val * val;
                }
                smB[t] += s; sqB[t] += q;
            }
        }
    }

#pragma unroll
    for (int t = 0; t < 4; ++t) {
        const int n = t * 16 + r16;
        atomicAdd(&lds_stats[n],      smA[t]);
        atomicAdd(&lds_stats[64 + n], sqA[t]);
        if (CONV) {
            atomicAdd(&lds_stats[128 + n], smB[t]);
            atomicAdd(&lds_stats[192 + n], sqB[t]);
        }
    }
    __syncthreads();
    for (int i = tid; i < NSTAT; i += 256) atomicAdd(&stats[i], lds_stats[i]);
}

// ---------------------------------------------------------------------------
// Fold BN stats + affine into per-feature scale/shift (bias-before-BN cancels).
// ---------------------------------------------------------------------------
__global__ void finalize_kernel(const float* __restrict__ stats,
                                const float* __restrict__ g,
                                const float* __restrict__ be,
                                float inv_cnt,
                                float* __restrict__ scale,
                                float* __restrict__ shift)
{
    int i = threadIdx.x;  // 64 threads
    float mean = stats[i] * inv_cnt;
    float var  = fmaxf(stats[64 + i] * inv_cnt - mean * mean, 0.f);
    float sc   = g[i] * rsqrtf(var + BN_EPS);
    scale[i] = sc;
    shift[i] = be[i] - mean * sc;
}

// v = silu(y*scale + shift)
__global__ void emb_act_kernel(const float* __restrict__ y,
                               const float* __restrict__ scale,
                               const float* __restrict__ shift,
                               float* __restrict__ v, int n)
{
    int i = blockIdx.x * blockDim.x + threadIdx.x;
    if (i >= n) return;
    int c = i & 63;
    v[i] = siluf_(y[i] * scale[c] + shift[c]);
}

// gated message + scatter-sum to destination nodes
__global__ __launch_bounds__(256) void edge_apply_kernel(
    const float* __restrict__ ym, const float* __restrict__ ys,
    const int* __restrict__ dst,
    const float* __restrict__ scM, const float* __restrict__ shM,
    const float* __restrict__ scS, const float* __restrict__ shS,
    float* __restrict__ agg, int E)
{
    int idx = blockIdx.x * 256 + threadIdx.x;
    if (idx >= E * 16) return;
    int e  = idx >> 4;
    int n0 = (idx & 15) * 4;
    const int d = dst[e];
    const float* pm = ym + (size_t)e * 64 + n0;
    const float* ps = ys + (size_t)e * 64 + n0;
    float* pa = agg + (size_t)d * 64 + n0;
#pragma unroll
    for (int j = 0; j < 4; ++j) {
        float m = sigmoidf_(pm[j] * scM[n0 + j] + shM[n0 + j]);
        float s = softplusf_(ps[j] * scS[n0 + j] + shS[n0 + j]);
        atomicAdd(&pa[j], m * s);
    }
}

// column stats (sum, sumsq per 64 features) over [R,64] matrix
__global__ void col_stats_kernel(const float* __restrict__ x, int total,
                                 float* __restrict__ stats)
{
    int tid0   = blockIdx.x * blockDim.x + threadIdx.x;
    int stride = gridDim.x * blockDim.x;   // launched as multiple of 64
    int c = tid0 & 63;
    float s = 0.f, q = 0.f;
    for (int i = tid0; i < total; i += stride) {
        float v = x[i];
        s += v; q += v * v;
    }
    atomicAdd(&stats[c], s);
    atomicAdd(&stats[64 + c], q);
}

// v = softplus(bn(agg) + v)
__global__ void node_update_kernel(const float* __restrict__ agg,
                                   const float* __restrict__ scale,
                                   const float* __restrict__ shift,
                                   float* __restrict__ v, int n)
{
    int i = blockIdx.x * blockDim.x + threadIdx.x;
    if (i >= n) return;
    int c = i & 63;
    v[i] = softplusf_(agg[i] * scale[c] + shift[c] + v[i]);
}

// per-graph sums + counts
__global__ void pool_kernel(const float* __restrict__ v,
                            const int* __restrict__ gids,
                            float* __restrict__ pooled,
                            float* __restrict__ counts, int N)
{
    int idx = blockIdx.x * blockDim.x + threadIdx.x;
    if (idx >= N * 16) return;
    int node = idx >> 4;
    int n0   = (idx & 15) * 4;
    int g = gids[node];
    const float* pv = v + (size_t)node * 64 + n0;
    float* pp = pooled + (size_t)g * 64 + n0;
#pragma unroll
    for (int j = 0; j < 4; ++j) atomicAdd(&pp[j], pv[j]);
    if ((idx & 15) == 0) atomicAdd(&counts[g], 1.f);
}

// Tail: vc = pooled/count -> fc (64->128) -> BN(100 rows) -> silu -> head (128->1)
__global__ __launch_bounds__(256) void tail_kernel(
    const float* __restrict__ pooled, const float* __restrict__ counts,
    const float* __restrict__ W_fc,   // [64,128]
    const float* __restrict__ g_fc, const float* __restrict__ be_fc,
    const float* __restrict__ W_out,  // [128,1]
    const float* __restrict__ b_out,
    float* __restrict__ out)          // [100]
{
    __shared__ float vc[100 * 64];
    __shared__ float ps[2][128], pq[2][128];
    __shared__ float sc[128], sh[128];
    __shared__ float outl[100];

    const int tid = threadIdx.x;
    for (int i = tid; i < 6400; i += 256) {
        int g = i >> 6;
        vc[i] = pooled[i] / fmaxf(counts[g], 1.f);
    }
    if (tid < 100) outl[tid] = 0.f;
    __syncthreads();

    const int c    = tid & 127;
    const int half = tid >> 7;
    const int r0   = half * 50;
    float h[50];
    float s = 0.f, q = 0.f;
    for (int j = 0; j < 50; ++j) {
        const float* vr = &vc[(r0 + j) * 64];
        float acc = 0.f;
#pragma unroll 8
        for (int k = 0; k < 64; ++k) acc += vr[k] * W_fc[k * 128 + c];
        h[j] = acc; s += acc; q += acc * acc;
    }
    ps[half][c] = s; pq[half][c] = q;
    __syncthreads();

    if (tid < 128) {
        float sum = ps[0][tid] + ps[1][tid];
        float sq  = pq[0][tid] + pq[1][tid];
        float mean = sum * 0.01f;
        float var  = fmaxf(sq * 0.01f - mean * mean, 0.f);
        float scl  = g_fc[tid] * rsqrtf(var + BN_EPS);
        sc[tid] = scl;
        sh[tid] = be_fc[tid] - mean * scl;
    }
    __syncthreads();

    const float wo = W_out[c];
    const float scl = sc[c], shf = sh[c];
    for (int j = 0; j < 50; ++j) {
        float y = siluf_(h[j] * scl + shf);
        atomicAdd(&outl[r0 + j], y * wo);
    }
    __syncthreads();
    if (tid < 100) out[tid] = outl[tid] + b_out[0];
}

// ---------------------------------------------------------------------------
extern "C" void kernel_launch(void* const* d_in, const int* in_sizes, int n_in,
                              void* d_out, int out_size, void* d_ws, size_t ws_size,
                              hipStream_t stream)
{
    (void)n_in; (void)ws_size;
    const float* node_feats = (const float*)d_in[0];
    const float* edge_feats = (const float*)d_in[1];
    const int*   src        = (const int*)d_in[2];
    const int*   dst        = (const int*)d_in[3];
    const int*   gids       = (const int*)d_in[4];
    const float* W_emb      = (const float*)d_in[5];
    const float* g_emb      = (const float*)d_in[7];
    const float* be_emb     = (const float*)d_in[8];
    const float* conv_Wm    = (const float*)d_in[9];
    const float* conv_gm    = (const float*)d_in[11];
    const float* conv_bem   = (const float*)d_in[12];
    const float* conv_Ws    = (const float*)d_in[13];
    const float* conv_gs    = (const float*)d_in[15];
    const float* conv_bes   = (const float*)d_in[16];
    const float* conv_gn    = (const float*)d_in[17];
    const float* conv_ben   = (const float*)d_in[18];
    const float* W_fc       = (const float*)d_in[19];
    const float* g_fc       = (const float*)d_in[21];
    const float* be_fc      = (const float*)d_in[22];
    const float* W_out      = (const float*)d_in[23];
    const float* b_out      = (const float*)d_in[24];

    const int N = in_sizes[0] / 92;
    const int E = in_sizes[2];
    const int C = in_sizes[9] / (169 * 64);
    const int G = out_size;

    float* w      = (float*)d_ws;
    float* v_buf  = w;
    float* agg    = v_buf + (size_t)N * 64;
    float* ym     = agg + (size_t)N * 64;
    float* ys     = ym + (size_t)E * 64;
    float* stats  = ys + (size_t)E * 64;
    float* scM    = stats + 256;
    float* shM    = scM + 64;
    float* scS    = shM + 64;
    float* shS    = scS + 64;
    float* scN    = shS + 64;
    float* shN    = scN + 64;
    float* pooled = shN + 64;
    float* counts = pooled + (size_t)G * 64;

    // ---- node embedding: Linear + BN + SiLU ----
    zero_kernel<<<1, 256, 0, stream>>>(stats, 128);
    gemm_wmma_kernel<0><<<512, 256, 0, stream>>>(
        node_feats, nullptr, nullptr, nullptr, W_emb, nullptr, ym, nullptr, stats, N);
    finalize_kernel<<<1, 64, 0, stream>>>(stats, g_emb, be_emb, 1.f / (float)N, scM, shM);
    emb_act_kernel<<<(N * 64 + 255) / 256, 256, 0, stream>>>(ym, scM, shM, v_buf, N * 64);

    // ---- conv layers ----
    for (int l = 0; l < C; ++l) {
        zero_kernel<<<1, 256, 0, stream>>>(stats, 256);
        gemm_wmma_kernel<1><<<512, 256, 0, stream>>>(
            v_buf, edge_feats, src, dst,
            conv_Wm + (size_t)l * 169 * 64, conv_Ws + (size_t)l * 169 * 64,
            ym, ys, stats, E);
        finalize_kernel<<<1, 64, 0, stream>>>(stats,       conv_gm + l * 64, conv_bem + l * 64,
                                              1.f / (float)E, scM, shM);
        finalize_kernel<<<1, 64, 0, stream>>>(stats + 128, conv_gs + l * 64, conv_bes + l * 64,
                                              1.f / (float)E, scS, shS);
        zero_kernel<<<(N * 64 + 255) / 256, 256, 0, stream>>>(agg, N * 64);
        edge_apply_kernel<<<(E * 16 + 255) / 256, 256, 0, stream>>>(
            ym, ys, dst, scM, shM, scS, shS, agg, E);
        zero_kernel<<<1, 256, 0, stream>>>(stats, 128);
        col_stats_kernel<<<64, 256, 0, stream>>>(agg, N * 64, stats);
        finalize_kernel<<<1, 64, 0, stream>>>(stats, conv_gn + l * 64, conv_ben + l * 64,
                                              1.f / (float)N, scN, shN);
        node_update_kernel<<<(N * 64 + 255) / 256, 256, 0, stream>>>(agg, scN, shN, v_buf, N * 64);
    }

    // ---- pooling + fc + head ----
    zero_kernel<<<(G * 64 + G + 255) / 256, 256, 0, stream>>>(pooled, G * 64 + G);
    pool_kernel<<<(N * 16 + 255) / 256, 256, 0, stream>>>(v_buf, gids, pooled, counts, N);
    tail_kernel<<<1, 256, 0, stream>>>(pooled, counts, W_fc, g_fc, be_fc, W_out, b_out,
                                       (float*)d_out);
}